// TensorTrainGaussian2D_51616916963915
// MI455X (gfx1250) — compile-verified
//
#include <hip/hip_runtime.h>
#include <hip/hip_bf16.h>

// TensorTrain Gaussian 2D log-likelihood for MI455X (gfx1250, wave32).
// arg tiles computed on the matrix pipe via V_WMMA_F32_16X16X4_F32;
// exp2 on the trans unit; cross-lane inner[] gather via ds_bpermute (__shfl).
// B-matrix operands fetched as packed {q,p}/{r,0} pairs -> one ds_load_b64,
// no divergent/exec-masked LDS loads in the hot loop.

typedef __attribute__((ext_vector_type(2))) float v2f;
typedef __attribute__((ext_vector_type(8))) float v8f;

#define KDIM 32
#define TBL (KDIM * KDIM)           // 1024
// float offsets into workspace / LDS
#define WS_QP 0                     // 2048 floats: {q,p} pairs (lanes 0-15)
#define WS_RZ (2 * TBL)             // 2048 floats: {r,0} pairs (lanes 16-31)
#define WS_VA (4 * TBL)             // 1024: stage-2 weights (w10 path)
#define WS_VB (5 * TBL)             // 1024: stage-1 weights (w21 path)
#define WS_W0 (6 * TBL)             // 32
#define WS_TOTAL (6 * TBL + KDIM)   // 6176 floats (~24.7 KB)

#define LOG2E 1.4426950408889634f
#define LN2   0.6931471805599453f
#define INV_SQRT_2PI 0.3989422804014327f
#define EPSF  2.220446049250313e-16f

__device__ __forceinline__ float fast_exp2(float x) {
#if __has_builtin(__builtin_amdgcn_exp2f)
  return __builtin_amdgcn_exp2f(x);          // v_exp_f32
#else
  return exp2f(x);
#endif
}
__device__ __forceinline__ float fast_ln(float x) {
#if __has_builtin(__builtin_amdgcn_logf)
  return LN2 * __builtin_amdgcn_logf(x);     // v_log_f32 * ln2
#else
  return logf(x);
#endif
}

__device__ __forceinline__ v8f wmma_16x16x4_f32(v2f a, v2f b, v8f c) {
#if __has_builtin(__builtin_amdgcn_wmma_f32_16x16x4_f32)
  // 8 args: (neg_a, A, neg_b, B, c_mod, C, reuse_a, reuse_b)
  return __builtin_amdgcn_wmma_f32_16x16x4_f32(false, a, false, b, (short)0, c,
                                               false, false);
#else
  v8f d;
  asm volatile("v_wmma_f32_16x16x4_f32 %0, %1, %2, %3"
               : "=v"(d) : "v"(a), "v"(b), "v"(c));
  return d;
#endif
}

// ---------------------------------------------------------------------------
// Prep: softmaxes + fused per-(a,b) coefficients. One wave, column per lane.
// ---------------------------------------------------------------------------
__global__ void ttg_prep(const float* __restrict__ Wk0,
                         const float* __restrict__ W10,
                         const float* __restrict__ W21,
                         const float* __restrict__ mu,
                         const float* __restrict__ sigma,
                         float* __restrict__ ws) {
  const int b = threadIdx.x;  // 0..31

  // softmax(Wk0) via wave reductions
  float v = Wk0[b];
  float m = v;
  for (int msk = 16; msk; msk >>= 1) m = fmaxf(m, __shfl_xor(m, msk, 32));
  float e = __expf(v - m);
  float s = e;
  for (int msk = 16; msk; msk >>= 1) s += __shfl_xor(s, msk, 32);
  ws[WS_W0 + b] = e / s;

  // column-wise (axis=0) softmax for W10 / W21: lane owns column b
  float m10 = -1e30f, m21 = -1e30f;
  for (int a = 0; a < KDIM; ++a) {
    m10 = fmaxf(m10, W10[a * KDIM + b]);
    m21 = fmaxf(m21, W21[a * KDIM + b]);
  }
  float s10 = 0.f, s21 = 0.f;
  for (int a = 0; a < KDIM; ++a) {
    s10 += __expf(W10[a * KDIM + b] - m10);
    s21 += __expf(W21[a * KDIM + b] - m21);
  }
  for (int a = 0; a < KDIM; ++a) {
    const int idx = a * KDIM + b;
    const float w10 = __expf(W10[idx] - m10) / s10;
    const float w21 = __expf(W21[idx] - m21) / s21;
    const float muv = mu[idx];
    const float inv_s = 1.0f / sigma[idx];
    // exp2 domain: arg = q*x^2 + p*x + r  with q = -0.5*log2e/sigma^2
    const float q = -0.5f * LOG2E * inv_s * inv_s;
    ws[WS_QP + 2 * idx + 0] = q;
    ws[WS_QP + 2 * idx + 1] = -2.0f * q * muv;
    ws[WS_RZ + 2 * idx + 0] = q * muv * muv;
    ws[WS_RZ + 2 * idx + 1] = 0.0f;
    ws[WS_VA + idx] = w10 * INV_SQRT_2PI * inv_s;
    ws[WS_VB + idx] = w21 * INV_SQRT_2PI * inv_s;
  }
}

// ---------------------------------------------------------------------------
// Main: one wave per 16-sample tile.
// ---------------------------------------------------------------------------
__global__ __launch_bounds__(256) void ttg_main(const float* __restrict__ X,
                                                const float* __restrict__ ws,
                                                float* __restrict__ out,
                                                int nTiles, int N) {
  __shared__ float smem[WS_TOTAL];
  for (int i = threadIdx.x; i < WS_TOTAL; i += blockDim.x) smem[i] = ws[i];
  __syncthreads();  // once, before the persistent loop (uniform)

  const float* s_va = smem + WS_VA;
  const float* s_vb = smem + WS_VB;
  const float* s_w0 = smem + WS_W0;

  const int lane  = threadIdx.x & 31;
  const int col0  = lane & 15;
  const bool hi   = lane >= 16;
  // per-lane, loop-invariant base for the packed B pairs: {q,p} or {r,0}
  const float* pairBase = hi ? (smem + WS_RZ) : (smem + WS_QP);

  const int gwave = blockIdx.x * (blockDim.x >> 5) + (threadIdx.x >> 5);
  const int nwave = gridDim.x * (blockDim.x >> 5);
  const float2* __restrict__ X2 = (const float2*)X;

  for (int tile = gwave; tile < nTiles; tile += nwave) {
    const int base = tile * 16;
    const int nrow = min(base + col0, N - 1);
    const float2 xv = X2[nrow];           // one global_load_b64
    const float x0 = xv.x, x1 = xv.y;

    // A (16x4 f32): lanes 0-15 -> K={0,1}={x^2,x}; lanes 16-31 -> K={2,3}={1,0}
    v2f A0, A1;
    A0[0] = hi ? 1.0f : x0 * x0;  A0[1] = hi ? 0.0f : x0;
    A1[0] = hi ? 1.0f : x1 * x1;  A1[1] = hi ? 0.0f : x1;

    // ---- stage 1: inner[n,b] = sum_a v21[a,b] * pdf1[n,a,b] -------------
    float innerAcc[2][8];
#pragma unroll
    for (int h = 0; h < 2; ++h)
#pragma unroll
      for (int g = 0; g < 8; ++g) innerAcc[h][g] = 0.0f;

#pragma unroll 2
    for (int a = 0; a < KDIM; ++a) {
#pragma unroll
      for (int h = 0; h < 2; ++h) {
        const int idx = a * KDIM + col0 + 16 * h;
        const v2f B = *(const v2f*)(pairBase + 2 * idx);  // ds_load_b64
        v8f c = {};
        v8f d = wmma_16x16x4_f32(A1, B, c);
        const float w = s_vb[idx];
#pragma unroll
        for (int g = 0; g < 8; ++g)
          innerAcc[h][g] = fmaf(w, fast_exp2(d[g]), innerAcc[h][g]);
      }
    }

    // ---- stage 2: mid[n,b] = sum_a vA[a,b] * pdf0[n,a,b] * inner[n,a] ---
    float midAcc[2][8];
#pragma unroll
    for (int h = 0; h < 2; ++h)
#pragma unroll
      for (int g = 0; g < 8; ++g) midAcc[h][g] = 0.0f;

#pragma unroll
    for (int aset = 0; aset < 2; ++aset) {   // static index into innerAcc
#pragma unroll 2
      for (int a16 = 0; a16 < 16; ++a16) {
        const int a = aset * 16 + a16;
        // inner[n_l, a] lives at lane (a&15)|(l&16), reg innerAcc[aset][g]
        const int srcLane = a16 | (lane & 16);
        float iv[8];
#pragma unroll
        for (int g = 0; g < 8; ++g)
          iv[g] = __shfl(innerAcc[aset][g], srcLane, 32);  // ds_bpermute
#pragma unroll
        for (int h = 0; h < 2; ++h) {
          const int idx = a * KDIM + col0 + 16 * h;
          const v2f B = *(const v2f*)(pairBase + 2 * idx);  // ds_load_b64
          v8f c = {};
          v8f d = wmma_16x16x4_f32(A0, B, c);
          const float w = s_va[idx];
#pragma unroll
          for (int g = 0; g < 8; ++g)
            midAcc[h][g] = fmaf(w * iv[g], fast_exp2(d[g]), midAcc[h][g]);
        }
      }
    }

    // ---- stage 3: lik[n] = sum_b mid[n,b]*w0[b]; log; write ------------
    const float w0a = s_w0[col0];
    const float w0b = s_w0[col0 + 16];
#pragma unroll
    for (int g = 0; g < 8; ++g) {
      float p = midAcc[0][g] * w0a + midAcc[1][g] * w0b;
      p += __shfl_xor(p, 1, 32);   // butterfly within 16-lane half
      p += __shfl_xor(p, 2, 32);
      p += __shfl_xor(p, 4, 32);
      p += __shfl_xor(p, 8, 32);
      if (col0 == 0) {             // lanes 0 and 16 own rows {0..7},{8..15}
        const int n2 = base + (hi ? 8 : 0) + g;
        if (n2 < N) out[n2] = fast_ln(p + EPSF);
      }
    }
  }
}

extern "C" void kernel_launch(void* const* d_in, const int* in_sizes, int n_in,
                              void* d_out, int out_size, void* d_ws, size_t ws_size,
                              hipStream_t stream) {
  (void)in_sizes; (void)n_in; (void)ws_size;
  const float* X     = (const float*)d_in[0];
  const float* Wk0   = (const float*)d_in[1];
  const float* W10   = (const float*)d_in[2];
  const float* W21   = (const float*)d_in[3];
  const float* mu    = (const float*)d_in[4];
  const float* sigma = (const float*)d_in[5];
  float* out = (float*)d_out;
  float* ws  = (float*)d_ws;   // needs 6176 floats (~24.7 KB)

  const int N = out_size;                       // 100000
  ttg_prep<<<1, 32, 0, stream>>>(Wk0, W10, W21, mu, sigma, ws);

  const int nTiles = (N + 15) / 16;             // 6250
  const int wavesPerBlock = 8;                  // 256 threads
  int blocks = (nTiles + wavesPerBlock - 1) / wavesPerBlock;
  if (blocks > 4096) blocks = 4096;
  ttg_main<<<blocks, 256, 0, stream>>>(X, ws, out, nTiles, N);
}